// GCNEncoder_11166914970268
// MI455X (gfx1250) — compile-verified
//
#include <hip/hip_runtime.h>

typedef __attribute__((ext_vector_type(2))) float v2f;
typedef __attribute__((ext_vector_type(8))) float v8f;

// ---------------------------------------------------------------- degree ----
__global__ void k_deg_init(float* __restrict__ deg, int n) {
  int i = blockIdx.x * blockDim.x + threadIdx.x;
  if (i < n) deg[i] = 1.0f;  // self-loop contributes 1 to every node's degree
}

__global__ void k_deg_edges(const int* __restrict__ dst, float* __restrict__ deg, int E) {
  int e = blockIdx.x * blockDim.x + threadIdx.x;
  if (e < E) unsafeAtomicAdd(&deg[dst[e]], 1.0f);
}

__global__ void k_dinv(float* __restrict__ deg, int n) {
  int i = blockIdx.x * blockDim.x + threadIdx.x;
  if (i < n) deg[i] = rsqrtf(deg[i]);  // deg >= 1 always (self-loops)
}

// ------------------------------------------------------- layer-1 tiny GEMM --
// h1[n, hid] = x[n, in_c] @ W1[in_c, hid]; K = 5 -> plain VALU, one thread per
// output element. x row is reused 128x from L0; W1 (2.5 KB) lives in cache.
__global__ void k_gemm1(const float* __restrict__ x, const float* __restrict__ W1,
                        float* __restrict__ h1, int n, int in_c, int hid) {
  int idx = blockIdx.x * blockDim.x + threadIdx.x;
  if (idx >= n * hid) return;
  int node = idx / hid;
  int c = idx - node * hid;
  float acc = 0.0f;
  for (int k = 0; k < in_c; ++k) acc += x[node * in_c + k] * W1[k * hid + c];
  h1[idx] = acc;
}

// ---------------------------------------- self-loop term + bias (agg init) --
// out[i,c] = h[i,c] * dinv[i]^2 + b[c]   (fully initializes the agg buffer,
// so the edge-scatter kernel only needs atomic adds on top)
__global__ void k_selfbias(const float* __restrict__ h, const float* __restrict__ dinv,
                           const float* __restrict__ b, float* __restrict__ out,
                           int n, int C) {
  int idx = blockIdx.x * blockDim.x + threadIdx.x;
  if (idx >= n * C) return;
  int node = idx / C;
  int c = idx - node * C;
  float di = dinv[node];
  out[idx] = h[idx] * di * di + b[c];
}

// --------------------------------------------------- edge scatter, C = 128 --
// One wave32 per edge. Lane l covers channels [4l, 4l+4): one float4 gather of
// h[src], scale by dinv[src]*dinv[dst], 4 hardware f32 atomic adds to out[dst].
__global__ void k_scatter128(const float* __restrict__ h, const float* __restrict__ dinv,
                             const int* __restrict__ src, const int* __restrict__ dst,
                             float* __restrict__ out, int E) {
  int gid = blockIdx.x * blockDim.x + threadIdx.x;
  int e = gid >> 5;
  int lane = gid & 31;
  if (e >= E) return;
  int s = src[e];
  int d = dst[e];
  float norm = dinv[s] * dinv[d];
  const float4 v = *(const float4*)(h + (size_t)s * 128 + lane * 4);
  float* op = out + (size_t)d * 128 + lane * 4;
  unsafeAtomicAdd(op + 0, v.x * norm);
  unsafeAtomicAdd(op + 1, v.y * norm);
  unsafeAtomicAdd(op + 2, v.z * norm);
  unsafeAtomicAdd(op + 3, v.w * norm);
}

// ---------------------------------------------------- edge scatter, C = 64 --
__global__ void k_scatter64(const float* __restrict__ h, const float* __restrict__ dinv,
                            const int* __restrict__ src, const int* __restrict__ dst,
                            float* __restrict__ out, int E) {
  int gid = blockIdx.x * blockDim.x + threadIdx.x;
  int e = gid >> 5;
  int lane = gid & 31;
  if (e >= E) return;
  int s = src[e];
  int d = dst[e];
  float norm = dinv[s] * dinv[d];
  const float2 v = *(const float2*)(h + (size_t)s * 64 + lane * 2);
  float* op = out + (size_t)d * 64 + lane * 2;
  unsafeAtomicAdd(op + 0, v.x * norm);
  unsafeAtomicAdd(op + 1, v.y * norm);
}

// ------------------------------------------------ layer-2 GEMM via f32 WMMA -
// D[n, out_c] = relu(A[n, hid]) @ B[hid, out_c], exact f32 via
// V_WMMA_F32_16X16X4_F32. One wave per 16x16 output tile, 32 k-steps of K=4.
//
// Per the ISA VGPR layouts (wave32):
//   A 16x4:  lanes 0-15 -> rows M=0..15, VGPR0=K, VGPR1=K+1
//            lanes 16-31 -> rows M=0..15, VGPR0=K+2, VGPR1=K+3
//   B 4x16:  lanes 0-15 -> cols N=0..15, VGPR0=row K, VGPR1=row K+1
//            lanes 16-31 -> cols N=0..15, VGPR0=row K+2, VGPR1=row K+3
//   C/D 16x16: VGPR r -> (M=r, N=lane) for lanes 0-15; (M=r+8, N=lane-16) else
__global__ void k_gemm2_wmma(const float* __restrict__ A, const float* __restrict__ B,
                             float* __restrict__ D, int n, int hid, int out_c) {
  int wid  = (blockIdx.x * blockDim.x + threadIdx.x) >> 5;  // uniform per wave
  int lane = threadIdx.x & 31;
  int nTiles = out_c >> 4;
  int mTile = wid / nTiles;
  int nTile = wid - mTile * nTiles;
  if (mTile * 16 >= n) return;  // wave-uniform exit: EXEC stays all-1s for WMMA

  int half = lane >> 4;   // 0: lanes 0-15, 1: lanes 16-31
  int l16  = lane & 15;

  int rowA = mTile * 16 + l16;
  if (rowA >= n) rowA = n - 1;            // clamp loads; stores are masked
  int colB = nTile * 16 + l16;

  const float* arow = A + (size_t)rowA * hid;
  v8f c = {};
  for (int k = 0; k < hid; k += 4) {
    int kk = k + half * 2;
    v2f a;
    a.x = fmaxf(arow[kk + 0], 0.0f);      // fused ReLU on A
    a.y = fmaxf(arow[kk + 1], 0.0f);
    const float* bp = B + (size_t)kk * out_c + colB;
    v2f b;
    b.x = bp[0];
    b.y = bp[out_c];
    // 8 args: (neg_a, A, neg_b, B, c_mod, C, reuse_a, reuse_b)
    c = __builtin_amdgcn_wmma_f32_16x16x4_f32(false, a, false, b, (short)0, c,
                                              false, false);
  }
#pragma unroll
  for (int r = 0; r < 8; ++r) {
    int row = mTile * 16 + r + half * 8;
    if (row < n) D[(size_t)row * out_c + nTile * 16 + l16] = c[r];
  }
}

// ---------------------------------------------------------------------------
extern "C" void kernel_launch(void* const* d_in, const int* in_sizes, int n_in,
                              void* d_out, int out_size, void* d_ws, size_t ws_size,
                              hipStream_t stream) {
  const float* x  = (const float*)d_in[0];
  const int*   ei = (const int*)d_in[1];
  const float* W1 = (const float*)d_in[2];
  const float* b1 = (const float*)d_in[3];
  const float* W2 = (const float*)d_in[4];
  const float* b2 = (const float*)d_in[5];
  float* out = (float*)d_out;

  int hid   = in_sizes[3];           // 128
  int out_c = in_sizes[5];           // 64
  int in_c  = in_sizes[2] / hid;     // 5
  int n     = in_sizes[0] / in_c;    // 100000
  int E     = in_sizes[1] / 2;       // 1600000

  const int* src = ei;
  const int* dst = ei + E;

  float* ws   = (float*)d_ws;
  float* dinv = ws;                          // n
  float* h1   = dinv + n;                    // n * hid (also reused as h2)
  float* agg1 = h1 + (size_t)n * hid;        // n * hid
  float* h2   = h1;                          // layer-2 GEMM output (n * out_c)

  const int B256 = 256;
  // degree -> dinv
  k_deg_init<<<(n + B256 - 1) / B256, B256, 0, stream>>>(dinv, n);
  k_deg_edges<<<(E + B256 - 1) / B256, B256, 0, stream>>>(dst, dinv, E);
  k_dinv<<<(n + B256 - 1) / B256, B256, 0, stream>>>(dinv, n);

  // layer 1
  int tot1 = n * hid;
  k_gemm1<<<(tot1 + B256 - 1) / B256, B256, 0, stream>>>(x, W1, h1, n, in_c, hid);
  k_selfbias<<<(tot1 + B256 - 1) / B256, B256, 0, stream>>>(h1, dinv, b1, agg1, n, hid);
  k_scatter128<<<(E + 7) / 8, B256, 0, stream>>>(h1, dinv, src, dst, agg1, E);

  // layer 2 (ReLU fused into WMMA A loads)
  int mTiles = (n + 15) / 16;
  int nTiles = out_c / 16;
  int waves  = mTiles * nTiles;
  k_gemm2_wmma<<<(waves + 7) / 8, B256, 0, stream>>>(agg1, W2, h2, n, hid, out_c);

  int tot2 = n * out_c;
  k_selfbias<<<(tot2 + B256 - 1) / B256, B256, 0, stream>>>(h2, dinv, b2, out, n, out_c);
  k_scatter64<<<(E + 7) / 8, B256, 0, stream>>>(h2, dinv, src, dst, out, E);
}